// signature_model_10746008175156
// MI455X (gfx1250) — compile-verified
//
#include <hip/hip_runtime.h>

typedef __attribute__((ext_vector_type(2))) float v2f;
typedef __attribute__((ext_vector_type(8))) float v8f;

#define LPATH    256
#define CDIM     8
#define SIGDIM   4680
#define NSTATE   584          // 8 + 64 + 512
#define U4STRIDE 66           // 64 K-slots + 2 pad: bank-friendly, 8B aligned
#define CHUNK    64
#define NOUT     10

// LDS float offsets
#define OFF_DXP  0                        // [256][16]  (cols 8..15 zero)  4096
#define OFF_SB   4096                     // [2][584]                      1168
#define OFF_U4   (4096 + 1168)            // [512][66]                    33792
#define OFF_S4   (OFF_U4 + 512*U4STRIDE)  // [512][8]                      4096
#define OFF_RED  (OFF_S4 + 4096)          // [10][256]                     2560
#define LDS_FLOATS (OFF_RED + NOUT*256)

__launch_bounds__(256, 1)
__global__ void sig_depth4_kernel(const float* __restrict__ X,
                                  const float* __restrict__ W,
                                  const float* __restrict__ bias,
                                  float* __restrict__ out) {
  extern __shared__ float lds[];
  float* dxp = lds + OFF_DXP;
  float* Sb  = lds + OFF_SB;
  float* U4  = lds + OFF_U4;
  float* S4s = lds + OFF_S4;
  float* red = lds + OFF_RED;

  const int b    = blockIdx.x;
  const int tid  = threadIdx.x;
  const int lane = tid & 31;
  const int wave = tid >> 5;
  const int lh   = lane >> 4;   // lane-half (0/1)
  const int ln   = lane & 15;

  // ---- Phase 0: load X[b], form dx rows (zero-padded to 16 cols; row 255 = 0 pad step)
  {
    const float* Xb = X + (size_t)b * (LPATH * CDIM);
    const int t = tid;  // 256 threads <-> 256 rows
    float4 d0 = make_float4(0.f, 0.f, 0.f, 0.f);
    float4 d1 = d0;
    if (t < LPATH - 1) {
      const float4* r0 = (const float4*)(Xb + (size_t)t * CDIM);
      const float4* r1 = (const float4*)(Xb + (size_t)(t + 1) * CDIM);
      float4 a0 = r0[0], a1 = r0[1], c0 = r1[0], c1 = r1[1];
      d0 = make_float4(c0.x - a0.x, c0.y - a0.y, c0.z - a0.z, c0.w - a0.w);
      d1 = make_float4(c1.x - a1.x, c1.y - a1.y, c1.z - a1.z, c1.w - a1.w);
    }
    float4* dr = (float4*)(dxp + t * 16);
    dr[0] = d0;
    dr[1] = d1;
    dr[2] = make_float4(0.f, 0.f, 0.f, 0.f);
    dr[3] = make_float4(0.f, 0.f, 0.f, 0.f);
  }
  // zero initial S1/S2/S3 state (buffer 0)
  for (int i = tid; i < NSTATE; i += 256) Sb[i] = 0.f;

  v8f acc0 = {}, acc1 = {}, acc2 = {}, acc3 = {};  // S4 rows for this wave
  __syncthreads();

  int cur = 0;
  for (int chunk = 0; chunk < 4; ++chunk) {
    const int t0 = chunk * CHUNK;

    // ---- Phase 1: sequential Chen scan (S1,S2,S3 double-buffered), record u4 into U4
    for (int s = 0; s < CHUNK; ++s) {
      const int t = t0 + s;
      const float* S1c = Sb + cur * NSTATE;
      const float* S2c = S1c + 8;
      const float* S3c = S1c + 72;
      float* S1n = Sb + (cur ^ 1) * NSTATE;
      float* S2n = S1n + 8;
      float* S3n = S1n + 72;
      const float* dx = dxp + t * 16;

      #pragma unroll
      for (int h = 0; h < 2; ++h) {
        const int beta = tid + h * 256;
        const int i = beta >> 6, j = (beta >> 3) & 7, k = beta & 7;
        const float dxi = dx[i], dxj = dx[j], dxk = dx[k];
        const float s1 = S1c[i], s2 = S2c[i * 8 + j], s3 = S3c[beta];
        // u4 = S3 + (S2/2 + (S1/6 + dx/24) (x) dx) (x) dx
        const float av = s1 * (1.f / 6.f) + dxi * (1.f / 24.f);
        const float bv = s2 * 0.5f + av * dxj;
        U4[beta * U4STRIDE + s] = s3 + bv * dxk;
        // S3 <- S3 + u3 (x) dx,  u3 = S2 + (S1/2 + dx/6) (x) dx
        const float u3 = s2 + (s1 * 0.5f + dxi * (1.f / 6.f)) * dxj;
        S3n[beta] = s3 + u3 * dxk;
      }
      if (tid < 64) {
        const int i = tid >> 3, j = tid & 7;
        S2n[tid] = S2c[tid] + (S1c[i] + dx[i] * 0.5f) * dx[j];  // u2 (x) dx
      }
      if (tid < 8) S1n[tid] = S1c[tid] + dx[tid];
      cur ^= 1;
      __syncthreads();
    }

    // ---- Phase 2: S4 += U4[512x64] * DX[64x8(pad16)] via v_wmma_f32_16x16x4_f32
    const int rowBase = wave * 64;  // each wave owns 4 tiles of 16 rows
    #pragma unroll
    for (int k0 = 0; k0 < CHUNK; k0 += 4) {
      const int kk = k0 + 2 * lh;  // K split across lane halves per ISA layout
      v2f bm;
      bm.x = dxp[(t0 + kk) * 16 + ln];       // B[K=kk][N=ln]
      bm.y = dxp[(t0 + kk + 1) * 16 + ln];   // B[K=kk+1][N=ln]
      const float* abase = U4 + (size_t)ln * U4STRIDE + kk;
      v2f a0 = *(const v2f*)(abase + (rowBase +  0) * U4STRIDE);
      v2f a1 = *(const v2f*)(abase + (rowBase + 16) * U4STRIDE);
      v2f a2 = *(const v2f*)(abase + (rowBase + 32) * U4STRIDE);
      v2f a3 = *(const v2f*)(abase + (rowBase + 48) * U4STRIDE);
      acc0 = __builtin_amdgcn_wmma_f32_16x16x4_f32(false, a0, false, bm, (short)0, acc0, false, false);
      acc1 = __builtin_amdgcn_wmma_f32_16x16x4_f32(false, a1, false, bm, (short)0, acc1, false, false);
      acc2 = __builtin_amdgcn_wmma_f32_16x16x4_f32(false, a2, false, bm, (short)0, acc2, false, false);
      acc3 = __builtin_amdgcn_wmma_f32_16x16x4_f32(false, a3, false, bm, (short)0, acc3, false, false);
    }
    __syncthreads();  // U4 fully consumed before next chunk overwrites it
  }

  // ---- Phase 3: spill S4 accumulators to LDS (C layout: VGPR v -> M = v + 8*lh, N = lane&15)
  {
    const int rowBase = wave * 64;
    if (ln < CDIM) {  // cols 8..15 are zero (B padding)
      #pragma unroll
      for (int v = 0; v < 8; ++v) {
        S4s[(rowBase +  0 + v + 8 * lh) * 8 + ln] = acc0[v];
        S4s[(rowBase + 16 + v + 8 * lh) * 8 + ln] = acc1[v];
        S4s[(rowBase + 32 + v + 8 * lh) * 8 + ln] = acc2[v];
        S4s[(rowBase + 48 + v + 8 * lh) * 8 + ln] = acc3[v];
      }
    }
  }
  __syncthreads();

  // ---- Phase 4: out[b,:] = W @ sig + bias ; sig = [S1|S2|S3|S4]
  const float* Sfin = Sb + cur * NSTATE;  // cur == 0 after 256 steps
  float part[NOUT];
  #pragma unroll
  for (int o = 0; o < NOUT; ++o) part[o] = 0.f;
  for (int idx = tid; idx < SIGDIM; idx += 256) {
    const float sv = (idx < NSTATE) ? Sfin[idx] : S4s[idx - NSTATE];
    const float* wp = W + idx;
    #pragma unroll
    for (int o = 0; o < NOUT; ++o) part[o] += sv * wp[(size_t)o * SIGDIM];
  }
  #pragma unroll
  for (int o = 0; o < NOUT; ++o) red[o * 256 + tid] = part[o];
  __syncthreads();
  for (int off = 128; off > 0; off >>= 1) {
    if (tid < off) {
      #pragma unroll
      for (int o = 0; o < NOUT; ++o) red[o * 256 + tid] += red[o * 256 + tid + off];
    }
    __syncthreads();
  }
  if (tid < NOUT) out[(size_t)b * NOUT + tid] = red[tid * 256] + bias[tid];
}

extern "C" void kernel_launch(void* const* d_in, const int* in_sizes, int n_in,
                              void* d_out, int out_size, void* d_ws, size_t ws_size,
                              hipStream_t stream) {
  (void)n_in; (void)out_size; (void)d_ws; (void)ws_size;
  const float* X    = (const float*)d_in[0];
  const float* W    = (const float*)d_in[1];
  const float* bias = (const float*)d_in[2];
  float* out = (float*)d_out;
  const int B = in_sizes[0] / (LPATH * CDIM);  // 128
  const size_t shmem = (size_t)LDS_FLOATS * sizeof(float);  // ~180 KB, fits 320 KB/WGP
  sig_depth4_kernel<<<dim3(B), dim3(256), shmem, stream>>>(X, W, bias, out);
}